// KernelConv1_11854109736984
// MI455X (gfx1250) — compile-verified
//
#include <hip/hip_runtime.h>

// Per-pixel adaptive 5x5 conv (KPN): out[b,n,y,x] = sum_{i,j} core[b,n,i*5+j,0,y,x] *
//                                                   frame[b,n,0,y+i-2,x+j-2]  (zero pad)
// Memory-bound (~453 MB moved for 210 MFLOP) -> optimize as a streaming kernel:
//  - core: coalesced b128 non-temporal loads (one-shot 419 MB stream, keep L2 for frames)
//  - frame: async global->LDS halo staging (CDNA5 GLOBAL_LOAD_ASYNC_TO_LDS, ASYNCcnt),
//           25-tap reuse served by ds_load_b128 from LDS.

typedef float v4f __attribute__((ext_vector_type(4)));

typedef __attribute__((address_space(1))) int* gptr_i32;
typedef __attribute__((address_space(3))) int* lptr_i32;

namespace {
constexpr int KK       = 5;
constexpr int PAD      = 2;
constexpr int TILE_H   = 16;
constexpr int TILE_W   = 64;                 // 16 threads * 4 px
constexpr int HALO_H   = TILE_H + 2 * PAD;   // 20
constexpr int HALO_W   = TILE_W + 2 * PAD;   // 68
constexpr int LDS_W    = 72;                 // padded row stride (floats), keeps b128 aligned
constexpr int NTHREADS = 256;                // 8 waves (wave32)
}

#if __has_builtin(__builtin_amdgcn_global_load_async_to_lds_b32)
#define USE_ASYNC_LDS 1
#endif

__global__ __launch_bounds__(NTHREADS) void kpn25_kernel(
    const float* __restrict__ frames,
    const float* __restrict__ core,
    float* __restrict__ out,
    int H, int W)
{
    __shared__ float lds[HALO_H * LDS_W];

    const int tid   = threadIdx.x;
    const int tx    = tid & 15;          // column group within tile (x4 floats)
    const int ty    = tid >> 4;          // row within tile
    const int tileX = blockIdx.x * TILE_W;
    const int tileY = blockIdx.y * TILE_H;
    const int plane = blockIdx.z;        // b*N + n  (C == 1)

    const long planeHW = (long)H * (long)W;
    const float* __restrict__ fplane = frames + (long)plane * planeHW;
    const float* __restrict__ cplane = core   + (long)plane * (long)(KK * KK) * planeHW;

    const int gx0 = tileX - PAD;
    const int gy0 = tileY - PAD;

    // ---- Stage frame halo (20x68, zero padded at borders) into LDS ----
    for (int e = tid; e < HALO_H * HALO_W; e += NTHREADS) {
        const int hr = e / HALO_W;
        const int hc = e - hr * HALO_W;
        const int gy = gy0 + hr;
        const int gx = gx0 + hc;
        const bool inb = ((unsigned)gy < (unsigned)H) && ((unsigned)gx < (unsigned)W);
#ifdef USE_ASYNC_LDS
        if (inb) {
            // Async DMA: global -> LDS without touching VGPRs; tracked by ASYNCcnt.
            __builtin_amdgcn_global_load_async_to_lds_b32(
                (gptr_i32)(fplane + (long)gy * W + gx),
                (lptr_i32)&lds[hr * LDS_W + hc],
                0, 0);
        } else {
            lds[hr * LDS_W + hc] = 0.0f;  // disjoint addresses from async writes -> no hazard
        }
#else
        lds[hr * LDS_W + hc] = inb ? fplane[(long)gy * W + gx] : 0.0f;
#endif
    }

#ifdef USE_ASYNC_LDS
#if __has_builtin(__builtin_amdgcn_s_wait_asynccnt)
    __builtin_amdgcn_s_wait_asynccnt(0);
#else
    asm volatile("s_wait_asynccnt 0x0" ::: "memory");
#endif
#endif
    __syncthreads();

    // ---- 25-tap weighted sum for a 1x4 pixel strip ----
    const int ox = tileX + tx * 4;
    const int oy = tileY + ty;
    const float* __restrict__ crow = cplane + (long)oy * W + ox;   // + tap*planeHW per tap
    const int lbase = ty * LDS_W + tx * 4;                          // halo(row ty+i, col tx*4+t)

    v4f acc = {0.0f, 0.0f, 0.0f, 0.0f};
#pragma unroll
    for (int i = 0; i < KK; ++i) {
        // 8 consecutive frame floats for this tap-row: two aligned ds_load_b128
        float f[8];
#pragma unroll
        for (int t = 0; t < 8; ++t)
            f[t] = lds[lbase + i * LDS_W + t];
#pragma unroll
        for (int j = 0; j < KK; ++j) {
            // core tap plane: coalesced 16B load, non-temporal (one-shot 419 MB stream)
            const v4f w = __builtin_nontemporal_load(
                (const v4f*)(crow + (long)(i * KK + j) * planeHW));
            acc.x += w.x * f[j + 0];
            acc.y += w.y * f[j + 1];
            acc.z += w.z * f[j + 2];
            acc.w += w.w * f[j + 3];
        }
    }

    __builtin_nontemporal_store(acc,
        (v4f*)(out + (long)plane * planeHW + (long)oy * W + ox));
}

extern "C" void kernel_launch(void* const* d_in, const int* in_sizes, int n_in,
                              void* d_out, int out_size, void* d_ws, size_t ws_size,
                              hipStream_t stream) {
    (void)n_in; (void)out_size; (void)d_ws; (void)ws_size;
    const float* frames = (const float*)d_in[0];   // [B,N,1,H,W] f32
    const float* core   = (const float*)d_in[1];   // [B,N,25,1,H,W] f32
    float*       out    = (float*)d_out;           // [B,N,H,W] f32

    const int H = 512, W = 512;
    const int nPlanes = in_sizes[0] / (H * W);     // B*N = 16

    dim3 grid(W / TILE_W, H / TILE_H, nPlanes);    // 8 x 32 x 16 = 4096 blocks
    kpn25_kernel<<<grid, NTHREADS, 0, stream>>>(frames, core, out, H, W);
}